// IFVDLoss_27358941675586
// MI455X (gfx1250) — compile-verified
//
#include <hip/hip_runtime.h>
#include <hip/hip_bf16.h>

typedef __attribute__((ext_vector_type(2))) float v2f;
typedef __attribute__((ext_vector_type(8))) float v8f;

#define B_   16
#define C_   19
#define H_   256
#define W_   256
#define HW_  (H_ * W_)
#define SH_  128          // source (preds_S) spatial size
#define TGT_ 512          // target label map size
#define NSUM 1024         // 32x32 padded sum tile per batch per tensor
#define SUMS_FLOATS (2 * B_ * NSUM + 2)

// ---------------------------------------------------------------------------
// On-the-fly bilinear sample of preds_S (128x128 -> 256x256, half-pixel,
// edge clamp == jax.image.resize 'bilinear' for upsampling).
// ---------------------------------------------------------------------------
__device__ __forceinline__ float bilinS(const float* __restrict__ S, int bc,
                                        int y, int x) {
    float sy = y * 0.5f - 0.25f;
    float sx = x * 0.5f - 0.25f;
    float fy0 = floorf(sy), fx0 = floorf(sx);
    int y0 = (int)fy0, x0 = (int)fx0;
    float fy = sy - fy0, fx = sx - fx0;
    int y0c = y0 < 0 ? 0 : y0;
    int y1c = (y0 + 1) > (SH_ - 1) ? (SH_ - 1) : (y0 + 1);
    int x0c = x0 < 0 ? 0 : x0;
    int x1c = (x0 + 1) > (SH_ - 1) ? (SH_ - 1) : (x0 + 1);
    const float* base = S + (size_t)bc * (SH_ * SH_);
    float v00 = base[y0c * SH_ + x0c];
    float v01 = base[y0c * SH_ + x1c];
    float v10 = base[y1c * SH_ + x0c];
    float v11 = base[y1c * SH_ + x1c];
    float top = v00 + fx * (v01 - v00);
    float bot = v10 + fx * (v11 - v10);
    return top + fy * (bot - top);
}

__global__ void k_zero(float* __restrict__ p, int n) {
    int i = blockIdx.x * blockDim.x + threadIdx.x;
    if (i < n) p[i] = 0.0f;
}

// ---------------------------------------------------------------------------
// Per-class sums via V_WMMA_F32_16X16X4_F32.
//   D(16x16) += A(16x4: channels x pixels) * B(4x16: onehot pixels x classes)
// A row 19 (mTile=1, local row 3) is forced to 1.0 for the S tensor so that
// sumsS[b][19][l] accumulates the class pixel count.
// Layout per ISA 7.12.2: lane L -> row/col = L%16; VGPR j -> K = 2*(L/16)+j.
// ---------------------------------------------------------------------------
#define SUM_BPB   8   // blocks per batch (gridDim.x)
#define SUM_WAVES 8   // 256 threads / 32

__global__ __launch_bounds__(256)
void k_sums(const float* __restrict__ S, const float* __restrict__ T,
            const int* __restrict__ target,
            float* __restrict__ sumsS, float* __restrict__ sumsT) {
    __shared__ float lsS[32 * 32];
    __shared__ float lsT[32 * 32];

    const int b   = blockIdx.y;
    const int tid = threadIdx.x;
    for (int i = tid; i < 1024; i += 256) { lsS[i] = 0.0f; lsT[i] = 0.0f; }
    __syncthreads();

    const int lane   = tid & 31;
    const int wave   = tid >> 5;
    const int laneM  = lane & 15;
    const int laneHi = lane >> 4;
    const int kA     = 2 * laneHi;   // K slot of VGPR j is kA + j

    v8f accS[2][2] = {};
    v8f accT[2][2] = {};

    const int nChunks = HW_ / 4;                      // 16384 (4 pixels/chunk)
    const int stride  = SUM_BPB * SUM_WAVES;          // uniform per wave
    const int tgtBase = b * TGT_ * TGT_;

    for (int ch = blockIdx.x * SUM_WAVES + wave; ch < nChunks; ch += stride) {
        const int pixBase = ch * 4;
        const int p0 = pixBase + kA;
        const int p1 = p0 + 1;
        const int y0p = p0 >> 8, x0p = p0 & 255;
        const int y1p = p1 >> 8, x1p = p1 & 255;

        const int lab0 = target[tgtBase + (2 * y0p) * TGT_ + 2 * x0p];
        const int lab1 = target[tgtBase + (2 * y1p) * TGT_ + 2 * x1p];

        // --- A fragments (channels x 4 pixels), two 16-row tiles ---
        v2f aS[2], aT[2];
        #pragma unroll
        for (int mt = 0; mt < 2; ++mt) {
            int c = mt * 16 + laneM;
            float s0, s1, t0, t1;
            if (c < C_) {
                int bc = b * C_ + c;
                s0 = bilinS(S, bc, y0p, x0p);
                s1 = bilinS(S, bc, y1p, x1p);
                const float* tb = T + (size_t)bc * HW_;
                t0 = tb[p0];
                t1 = tb[p1];
            } else if (c == C_) {           // ones-row -> class counts in sumsS
                s0 = 1.0f; s1 = 1.0f; t0 = 0.0f; t1 = 0.0f;
            } else {
                s0 = 0.0f; s1 = 0.0f; t0 = 0.0f; t1 = 0.0f;
            }
            aS[mt][0] = s0; aS[mt][1] = s1;
            aT[mt][0] = t0; aT[mt][1] = t1;
        }

        // --- B fragments (onehot: 4 pixels x classes), two 16-col tiles ---
        v2f bf[2];
        #pragma unroll
        for (int nt = 0; nt < 2; ++nt) {
            int cls = nt * 16 + laneM;
            bf[nt][0] = (lab0 == cls) ? 1.0f : 0.0f;
            bf[nt][1] = (lab1 == cls) ? 1.0f : 0.0f;
        }

        // --- WMMA accumulate (EXEC is all-1s here: reconverged, full waves) ---
        #pragma unroll
        for (int mt = 0; mt < 2; ++mt) {
            #pragma unroll
            for (int nt = 0; nt < 2; ++nt) {
                accS[mt][nt] = __builtin_amdgcn_wmma_f32_16x16x4_f32(
                    false, aS[mt], false, bf[nt], (short)0, accS[mt][nt],
                    false, false);
                accT[mt][nt] = __builtin_amdgcn_wmma_f32_16x16x4_f32(
                    false, aT[mt], false, bf[nt], (short)0, accT[mt][nt],
                    false, false);
            }
        }
    }

    // Merge wave accumulators into LDS (D layout: VGPR r -> M = r + 8*laneHi).
    #pragma unroll
    for (int mt = 0; mt < 2; ++mt) {
        #pragma unroll
        for (int nt = 0; nt < 2; ++nt) {
            #pragma unroll
            for (int r = 0; r < 8; ++r) {
                int row = mt * 16 + r + 8 * laneHi;
                int col = nt * 16 + laneM;
                atomicAdd(&lsS[row * 32 + col], accS[mt][nt][r]);
                atomicAdd(&lsT[row * 32 + col], accT[mt][nt][r]);
            }
        }
    }
    __syncthreads();

    for (int i = tid; i < 1024; i += 256) {
        int row = i >> 5, col = i & 31;
        if (row < C_ + 1 && col < C_) {    // rows 0..18 sums, row 19 counts
            atomicAdd(&sumsS[b * NSUM + i], lsS[i]);
            atomicAdd(&sumsT[b * NSUM + i], lsT[i]);
        }
    }
}

// ---------------------------------------------------------------------------
// Per-pixel pass: class means from sums -> LDS, then cosine-sim diff^2 and
// channel-softmax KLD, block-reduced into two global accumulators.
// ---------------------------------------------------------------------------
__global__ __launch_bounds__(256)
void k_loss(const float* __restrict__ S, const float* __restrict__ T,
            const int* __restrict__ target,
            const float* __restrict__ sumsS, const float* __restrict__ sumsT,
            float* __restrict__ gAcc) {
    __shared__ float lmS[C_ * C_];
    __shared__ float lmT[C_ * C_];
    __shared__ float red[256];

    const int b   = blockIdx.y;
    const int tid = threadIdx.x;

    for (int i = tid; i < C_ * C_; i += 256) {
        int c = i / C_, l = i % C_;
        float cnt = sumsS[b * NSUM + C_ * 32 + l];      // row 19 = counts
        float inv = 1.0f / (cnt + 1e-6f);
        lmS[i] = sumsS[b * NSUM + c * 32 + l] * inv;
        lmT[i] = sumsT[b * NSUM + c * 32 + l] * inv;
    }
    __syncthreads();

    float accSq = 0.0f, accKld = 0.0f;
    const int tgtBase = b * TGT_ * TGT_;

    for (int p = blockIdx.x * 256 + tid; p < HW_; p += gridDim.x * 256) {
        const int y = p >> 8, x = p & 255;
        const int lab = target[tgtBase + (2 * y) * TGT_ + 2 * x];

        float fS[C_], fT[C_];
        float maxS = -3.0e38f, maxT = -3.0e38f;
        #pragma unroll
        for (int c = 0; c < C_; ++c) {
            int bc = b * C_ + c;
            fS[c] = bilinS(S, bc, y, x);
            fT[c] = T[(size_t)bc * HW_ + p];
            maxS = fmaxf(maxS, fS[c]);
            maxT = fmaxf(maxT, fT[c]);
        }

        float dS = 0.f, nS = 0.f, nmS = 0.f;
        float dT = 0.f, nT = 0.f, nmT = 0.f;
        float eS = 0.f, eT = 0.f;
        #pragma unroll
        for (int c = 0; c < C_; ++c) {
            float ms = lmS[c * C_ + lab];
            float mt = lmT[c * C_ + lab];
            dS += fS[c] * ms;  nS += fS[c] * fS[c];  nmS += ms * ms;
            dT += fT[c] * mt;  nT += fT[c] * fT[c];  nmT += mt * mt;
            eS += __expf(fS[c] - maxS);
            eT += __expf(fT[c] - maxT);
        }
        float lseS = __logf(eS), lseT = __logf(eT);

        float cosS = dS / (fmaxf(sqrtf(nS), 1e-8f) * fmaxf(sqrtf(nmS), 1e-8f));
        float cosT = dT / (fmaxf(sqrtf(nT), 1e-8f) * fmaxf(sqrtf(nmT), 1e-8f));
        float diff = cosS - cosT;
        accSq += diff * diff;

        float kld = 0.0f;
        #pragma unroll
        for (int c = 0; c < C_; ++c) {
            float lpT = fT[c] - maxT - lseT;
            float lpS = fS[c] - maxS - lseS;
            kld += __expf(lpT) * (lpT - lpS);
        }
        accKld += kld;
    }

    // block tree reductions + one atomic each
    red[tid] = accSq;
    __syncthreads();
    #pragma unroll
    for (int s = 128; s > 0; s >>= 1) {
        if (tid < s) red[tid] += red[tid + s];
        __syncthreads();
    }
    if (tid == 0) atomicAdd(&gAcc[0], red[0]);
    __syncthreads();

    red[tid] = accKld;
    __syncthreads();
    #pragma unroll
    for (int s = 128; s > 0; s >>= 1) {
        if (tid < s) red[tid] += red[tid + s];
        __syncthreads();
    }
    if (tid == 0) atomicAdd(&gAcc[1], red[0]);
}

__global__ void k_final(const float* __restrict__ gAcc, float* __restrict__ out) {
    if (threadIdx.x == 0 && blockIdx.x == 0) {
        float lossPD   = gAcc[1] / ((float)B_ * (float)C_ * (float)H_);
        float lossIFVD = 10.0f * gAcc[0] / ((float)B_ * (float)H_ * (float)W_);
        out[0] = lossIFVD + lossPD;
    }
}

// ---------------------------------------------------------------------------
extern "C" void kernel_launch(void* const* d_in, const int* in_sizes, int n_in,
                              void* d_out, int out_size, void* d_ws, size_t ws_size,
                              hipStream_t stream) {
    const float* S      = (const float*)d_in[0];   // preds_S [16,19,128,128]
    const float* T      = (const float*)d_in[1];   // preds_T [16,19,256,256]
    const int*   target = (const int*)d_in[2];     // [16,1,512,512]
    // d_in[3] = step (unused by the loss math)

    float* sumsS = (float*)d_ws;                   // [B][32][32]
    float* sumsT = sumsS + B_ * NSUM;              // [B][32][32]
    float* gAcc  = sumsT + B_ * NSUM;              // [2]
    float* out   = (float*)d_out;

    k_zero<<<(SUMS_FLOATS + 255) / 256, 256, 0, stream>>>(sumsS, SUMS_FLOATS);

    dim3 gs(SUM_BPB, B_);
    k_sums<<<gs, 256, 0, stream>>>(S, T, target, sumsS, sumsT);

    dim3 gl(64, B_);
    k_loss<<<gl, 256, 0, stream>>>(S, T, target, sumsS, sumsT, gAcc);

    k_final<<<1, 32, 0, stream>>>(gAcc, out);
}